// PCACS_85177791415056
// MI455X (gfx1250) — compile-verified
//
#include <hip/hip_runtime.h>
#include <hip/hip_bf16.h>
#include <math.h>

typedef __attribute__((ext_vector_type(2))) float v2f;
typedef __attribute__((ext_vector_type(4))) float v4f;
typedef __attribute__((ext_vector_type(8))) float v8f;

#define BPI 32       // blocks per image (stats pass)
#define TPB 256      // threads per block
constexpr int Himg = 512, Wimg = 512;
constexpr int Npix = Himg * Wimg;   // 262144
constexpr int Bimg = 64;

__device__ __forceinline__ float wave_sum32(float v) {
  v += __shfl_xor(v, 16, 32);
  v += __shfl_xor(v, 8, 32);
  v += __shfl_xor(v, 4, 32);
  v += __shfl_xor(v, 2, 32);
  v += __shfl_xor(v, 1, 32);
  return v;
}

// ---------------------------------------------------------------------------
// Pass 1: per-block partial sums of [s0,s1,s2, s00,s01,s02,s11,s12,s22] on
// shifted data (x-128) for f32 conditioning. Per-wave shuffle reduce, then the
// 8x16 LDS tile of wave partials is collapsed with two V_WMMA_F32_16X16X4_F32
// ops (ones-matrix row reduction). Exact in f32.
// ---------------------------------------------------------------------------
__global__ __launch_bounds__(TPB) void pca_stats_kernel(
    const float* __restrict__ img, float* __restrict__ partials) {
  const int b = blockIdx.y;   // image
  const int g = blockIdx.x;   // slab within image
  const float* p0 = img + (size_t)b * 3 * Npix;
  const float* p1 = p0 + Npix;
  const float* p2 = p1 + Npix;

  float s0 = 0.f, s1 = 0.f, s2 = 0.f;
  float s00 = 0.f, s01 = 0.f, s02 = 0.f, s11 = 0.f, s12 = 0.f, s22 = 0.f;

  const int n4 = Npix / 4;          // 65536 float4 per plane
  const int stride4 = BPI * TPB;    // 8192
  for (int i4 = g * TPB + threadIdx.x; i4 < n4; i4 += stride4) {
    v4f x = ((const v4f*)p0)[i4];
    v4f y = ((const v4f*)p1)[i4];
    v4f z = ((const v4f*)p2)[i4];
#pragma unroll
    for (int e = 0; e < 4; ++e) {
      float xv = x[e] - 128.0f, yv = y[e] - 128.0f, zv = z[e] - 128.0f;
      s0 += xv; s1 += yv; s2 += zv;
      s00 = fmaf(xv, xv, s00); s01 = fmaf(xv, yv, s01);
      s02 = fmaf(xv, zv, s02); s11 = fmaf(yv, yv, s11);
      s12 = fmaf(yv, zv, s12); s22 = fmaf(zv, zv, s22);
    }
  }

  // in-wave reduction (wave32)
  s0 = wave_sum32(s0);  s1 = wave_sum32(s1);  s2 = wave_sum32(s2);
  s00 = wave_sum32(s00); s01 = wave_sum32(s01); s02 = wave_sum32(s02);
  s11 = wave_sum32(s11); s12 = wave_sum32(s12); s22 = wave_sum32(s22);

  __shared__ float lds[8 * 16];
  const int wave = threadIdx.x >> 5;
  const int lane = threadIdx.x & 31;
  if (lane == 0) {
    float* row = &lds[wave * 16];
    row[0] = s0;  row[1] = s1;  row[2] = s2;
    row[3] = s00; row[4] = s01; row[5] = s02;
    row[6] = s11; row[7] = s12; row[8] = s22;
#pragma unroll
    for (int n = 9; n < 16; ++n) row[n] = 0.f;
  }
  __syncthreads();

  if (threadIdx.x < 32) {  // wave 0, full EXEC (WMMA requirement)
    const int L = threadIdx.x;
    const int col = L & 15;
    const int rbase = (L >> 4) * 2;  // lanes 0-15 -> rows {0,1}, 16-31 -> {2,3}
    v2f a;  a.x = 1.0f; a.y = 1.0f;                 // all-ones A(16x4)
    v2f b0; b0.x = lds[(rbase + 0) * 16 + col];     // waves 0..3
            b0.y = lds[(rbase + 1) * 16 + col];
    v2f b1; b1.x = lds[(rbase + 4) * 16 + col];     // waves 4..7
            b1.y = lds[(rbase + 5) * 16 + col];
    v8f acc = {};
    acc = __builtin_amdgcn_wmma_f32_16x16x4_f32(false, a, false, b0,
                                                (short)0, acc, false, false);
    acc = __builtin_amdgcn_wmma_f32_16x16x4_f32(false, a, false, b1,
                                                (short)0, acc, false, false);
    // Row M=0 of D lives in element 0, lanes 0..15 = columns 0..15.
    if (L < 9) partials[((size_t)b * BPI + g) * 16 + L] = acc[0];
  }
}

// ---------------------------------------------------------------------------
// Pass 2: one thread per image. Sum the 32 block partials in fixed order
// (deterministic), build the 3x3 covariance, analytic symmetric eigensolve
// (eigvals sorted descending), eigvecs via cross products, then
// delta = V @ (alpha * lambda).
// ---------------------------------------------------------------------------
__device__ __forceinline__ void eigvec3(float a00, float a01, float a02,
                                        float a11, float a12, float a22,
                                        float l, float* vx, float* vy, float* vz) {
  float r0x = a00 - l, r0y = a01,     r0z = a02;
  float r1x = a01,     r1y = a11 - l, r1z = a12;
  float r2x = a02,     r2y = a12,     r2z = a22 - l;
  float c0x = r0y * r1z - r0z * r1y, c0y = r0z * r1x - r0x * r1z, c0z = r0x * r1y - r0y * r1x;
  float c1x = r0y * r2z - r0z * r2y, c1y = r0z * r2x - r0x * r2z, c1z = r0x * r2y - r0y * r2x;
  float c2x = r1y * r2z - r1z * r2y, c2y = r1z * r2x - r1x * r2z, c2z = r1x * r2y - r1y * r2x;
  float n0 = c0x * c0x + c0y * c0y + c0z * c0z;
  float n1 = c1x * c1x + c1y * c1y + c1z * c1z;
  float n2 = c2x * c2x + c2y * c2y + c2z * c2z;
  float bx = c0x, by = c0y, bz = c0z, bn = n0;
  if (n1 > bn) { bx = c1x; by = c1y; bz = c1z; bn = n1; }
  if (n2 > bn) { bx = c2x; by = c2y; bz = c2z; bn = n2; }
  float inv = rsqrtf(fmaxf(bn, 1e-30f));
  *vx = bx * inv; *vy = by * inv; *vz = bz * inv;
}

__global__ void pca_eigen_kernel(const float* __restrict__ partials,
                                 const float* __restrict__ alphas,
                                 float* __restrict__ delta) {
  const int b = threadIdx.x;
  if (b >= Bimg) return;
  float s[9];
#pragma unroll
  for (int n = 0; n < 9; ++n) s[n] = 0.f;
  for (int g = 0; g < BPI; ++g) {
    const float* row = partials + ((size_t)b * BPI + g) * 16;
#pragma unroll
    for (int n = 0; n < 9; ++n) s[n] += row[n];
  }
  const float Nf = (float)Npix;
  const float invN = 1.0f / Nf;
  const float sc = 1.0f / (Nf - 1.0f);
  float m0 = s[0] * invN, m1 = s[1] * invN, m2 = s[2] * invN;  // shifted means
  float a00 = (s[3] - s[0] * m0) * sc;
  float a01 = (s[4] - s[0] * m1) * sc;
  float a02 = (s[5] - s[0] * m2) * sc;
  float a11 = (s[6] - s[1] * m1) * sc;
  float a12 = (s[7] - s[1] * m2) * sc;
  float a22 = (s[8] - s[2] * m2) * sc;

  float w0, w1, w2;
  float V[3][3];  // columns = eigenvectors
  float p1 = a01 * a01 + a02 * a02 + a12 * a12;
  float q = (a00 + a11 + a22) * (1.0f / 3.0f);
  float b00 = a00 - q, b11 = a11 - q, b22 = a22 - q;
  float p2 = b00 * b00 + b11 * b11 + b22 * b22 + 2.0f * p1;
  float p = sqrtf(p2 * (1.0f / 6.0f));
  if (p < 1e-10f) {
    w0 = w1 = w2 = q;
    V[0][0] = 1.f; V[1][0] = 0.f; V[2][0] = 0.f;
    V[0][1] = 0.f; V[1][1] = 1.f; V[2][1] = 0.f;
    V[0][2] = 0.f; V[1][2] = 0.f; V[2][2] = 1.f;
  } else {
    float inv = 1.0f / p;
    float c00 = b00 * inv, c01 = a01 * inv, c02 = a02 * inv;
    float c11 = b11 * inv, c12 = a12 * inv, c22 = b22 * inv;
    float detB = c00 * (c11 * c22 - c12 * c12)
               - c01 * (c01 * c22 - c12 * c02)
               + c02 * (c01 * c12 - c11 * c02);
    float r = fminf(1.0f, fmaxf(-1.0f, 0.5f * detB));
    float phi = acosf(r) * (1.0f / 3.0f);
    w0 = q + 2.0f * p * cosf(phi);
    w2 = q + 2.0f * p * cosf(phi + 2.0943951023931953f);  // +2pi/3
    w1 = 3.0f * q - w0 - w2;                               // w0 >= w1 >= w2
    eigvec3(a00, a01, a02, a11, a12, a22, w0, &V[0][0], &V[1][0], &V[2][0]);
    eigvec3(a00, a01, a02, a11, a12, a22, w1, &V[0][1], &V[1][1], &V[2][1]);
    eigvec3(a00, a01, a02, a11, a12, a22, w2, &V[0][2], &V[1][2], &V[2][2]);
  }

  const float* al = alphas + b * 3;
  float g0 = al[0] * w0, g1 = al[1] * w1, g2 = al[2] * w2;
  delta[b * 3 + 0] = V[0][0] * g0 + V[0][1] * g1 + V[0][2] * g2;
  delta[b * 3 + 1] = V[1][0] * g0 + V[1][1] * g1 + V[1][2] * g2;
  delta[b * 3 + 2] = V[2][0] * g0 + V[2][1] * g1 + V[2][2] * g2;
}

// ---------------------------------------------------------------------------
// Pass 3: out = clip(x + delta[b,c], 0, 255), 128-bit vectors, non-temporal
// stores (input lines likely still resident in the 192MB L2 from pass 1).
// ---------------------------------------------------------------------------
__global__ __launch_bounds__(256) void pca_apply_kernel(
    const float* __restrict__ img, const float* __restrict__ delta,
    float* __restrict__ out) {
  const int plane = blockIdx.y;           // b*3 + c, 0..191
  const float d = delta[plane];
  const size_t base = (size_t)plane * Npix;
  const int i4 = blockIdx.x * blockDim.x + threadIdx.x;  // 0..65535
  const v4f* in4 = (const v4f*)(img + base);
  v4f* out4 = (v4f*)(out + base);
  v4f v = in4[i4];
#pragma unroll
  for (int e = 0; e < 4; ++e) v[e] = fminf(fmaxf(v[e] + d, 0.f), 255.f);
  __builtin_nontemporal_store(v, out4 + i4);
}

extern "C" void kernel_launch(void* const* d_in, const int* in_sizes, int n_in,
                              void* d_out, int out_size, void* d_ws, size_t ws_size,
                              hipStream_t stream) {
  const float* img = (const float*)d_in[0];      // (64,3,512,512) f32
  const float* alphas = (const float*)d_in[1];   // (64,3) f32
  float* out = (float*)d_out;
  float* partials = (float*)d_ws;                 // 64*32*16 f32 = 128 KB
  float* delta = partials + (size_t)Bimg * BPI * 16;  // 192 f32

  pca_stats_kernel<<<dim3(BPI, Bimg), TPB, 0, stream>>>(img, partials);
  pca_eigen_kernel<<<dim3(1), dim3(64), 0, stream>>>(partials, alphas, delta);
  pca_apply_kernel<<<dim3(Npix / 4 / 256, Bimg * 3), 256, 0, stream>>>(img, delta, out);
}